// MACEBlock_67577015435584
// MI455X (gfx1250) — compile-verified
//
#include <hip/hip_runtime.h>
#include <hip/hip_bf16.h>
#include <math.h>

typedef float v2f __attribute__((ext_vector_type(2)));
typedef float v8f __attribute__((ext_vector_type(8)));

__device__ __forceinline__ float silu_f(float x) {
  return x / (1.0f + __expf(-x));
}

// One 16x16 f32 output tile via V_WMMA_F32_16X16X4_F32, K = KCH*4.
// A[m][k] at A[m*a_stride + a_col0 + k*a_step]   (m = lane&15)
// B[k][n] at B[k*b_stride + b_col0 + n]          (n = lane&15)
// Layout per CDNA5 ISA 7.12.2: lane l, reg r -> K = 2*(l>>4) + r (A and B);
// C/D: reg r -> row r + 8*(l>>4), col = l&15.
template <int KCH>
__device__ __forceinline__ v8f wmma_tile(const float* A, int a_stride, int a_col0, int a_step,
                                         const float* B, int b_stride, int b_col0, v8f acc) {
  const int lane = threadIdx.x & 31;
  const int mn = lane & 15;
  const int half2 = (lane >> 4) << 1;  // 0 or 2
#pragma unroll
  for (int kk = 0; kk < KCH; ++kk) {
    const int k0 = (kk << 2) + half2;
    v2f a, b;
    a.x = A[mn * a_stride + a_col0 + (k0    ) * a_step];
    a.y = A[mn * a_stride + a_col0 + (k0 + 1) * a_step];
    b.x = B[(k0    ) * b_stride + b_col0 + mn];
    b.y = B[(k0 + 1) * b_stride + b_col0 + mn];
    acc = __builtin_amdgcn_wmma_f32_16x16x4_f32(false, a, false, b, (short)0, acc, false, false);
  }
  return acc;
}

__device__ __forceinline__ void store_tile(float* out, long row0, int o_col0, int o_step,
                                           int maxrow, v8f c) {
  const int lane = threadIdx.x & 31;
  const int n = lane & 15;
  const int half = lane >> 4;
#pragma unroll
  for (int r = 0; r < 8; ++r) {
    long row = row0 + r + half * 8;
    if (row < maxrow) out[row * 256 + o_col0 + (long)n * o_step] = c[r];
  }
}

// store a 16x16 D tile into LDS activation buffer (stride 64) with SiLU applied
__device__ __forceinline__ void store_act_silu(float* act, int col0, v8f c) {
  const int lane = threadIdx.x & 31;
  const int n = lane & 15;
  const int half = lane >> 4;
#pragma unroll
  for (int r = 0; r < 8; ++r)
    act[(r + half * 8) * 64 + col0 + n] = silu_f(c[r]);
}

// out[n, :] = [ in1_0 @ W0a  |  in1_x @ W1a interleaved x=0..2 ] (+ same with in2/W*b)
// Rows: 32 per block, layout per row: [0:64) scalar, col 64 + c*3 + x vector.
__global__ __launch_bounds__(256) void
dual_linear_kernel(const float* __restrict__ in1, const float* __restrict__ W0a,
                   const float* __restrict__ W1a, const float* __restrict__ in2,
                   const float* __restrict__ W0b, const float* __restrict__ W1b,
                   float* __restrict__ out, int N) {
  __shared__ float tile[32 * 256];
  const int tid = threadIdx.x;
  const long base = (long)blockIdx.x * 32;
  const int w = tid >> 5;

  v8f acc[4];
#pragma unroll
  for (int i = 0; i < 4; ++i) acc[i] = (v8f){0,0,0,0,0,0,0,0};

  for (int i = tid; i < 32 * 256; i += 256) {
    long row = base + (i >> 8);
    if (row >= N) row = N - 1;
    tile[i] = in1[row * 256 + (i & 255)];
  }
  __syncthreads();
#pragma unroll
  for (int i = 0; i < 4; ++i) {
    int t = w + 8 * i;
    int mat = t >> 3, rt = (t >> 2) & 1, ct = t & 3;
    const float* Wm = (mat == 0) ? W0a : W1a;
    int a_col0 = (mat == 0) ? 0 : (64 + (mat - 1));
    int a_step = (mat == 0) ? 1 : 3;
    acc[i] = wmma_tile<16>(tile + rt * 16 * 256, 256, a_col0, a_step, Wm, 64, ct * 16, acc[i]);
  }
  if (in2) {
    __syncthreads();
    for (int i = tid; i < 32 * 256; i += 256) {
      long row = base + (i >> 8);
      if (row >= N) row = N - 1;
      tile[i] = in2[row * 256 + (i & 255)];
    }
    __syncthreads();
#pragma unroll
    for (int i = 0; i < 4; ++i) {
      int t = w + 8 * i;
      int mat = t >> 3, rt = (t >> 2) & 1, ct = t & 3;
      const float* Wm = (mat == 0) ? W0b : W1b;
      int a_col0 = (mat == 0) ? 0 : (64 + (mat - 1));
      int a_step = (mat == 0) ? 1 : 3;
      acc[i] = wmma_tile<16>(tile + rt * 16 * 256, 256, a_col0, a_step, Wm, 64, ct * 16, acc[i]);
    }
  }
#pragma unroll
  for (int i = 0; i < 4; ++i) {
    int t = w + 8 * i;
    int mat = t >> 3, rt = (t >> 2) & 1, ct = t & 3;
    int o_col0 = (mat == 0) ? ct * 16 : (64 + (mat - 1) + ct * 48);
    int o_step = (mat == 0) ? 1 : 3;
    store_tile(out, base + rt * 16, o_col0, o_step, N, acc[i]);
  }
}

// Fused: RBF MLP (WMMA) -> tpw in registers -> gather up[snd] -> message -> atomic scatter msg[rcv]
__global__ __launch_bounds__(256) void
edge_kernel(const int* __restrict__ eidx, const float* __restrict__ esh,
            const float* __restrict__ erbf, const float* __restrict__ M1,
            const float* __restrict__ M2, const float* __restrict__ M3,
            const float* __restrict__ M4, const float* __restrict__ up,
            float* __restrict__ msg, int E) {
  __shared__ float rbf[32 * 16];
  __shared__ float actA[32 * 64];
  __shared__ float actB[32 * 64];
  const int tid = threadIdx.x;
  const long e0 = (long)blockIdx.x * 32;

  for (int i = tid; i < 32 * 16; i += 256) {
    long e = e0 + (i >> 4);
    if (e >= E) e = E - 1;
    rbf[i] = erbf[e * 16 + (i & 15)];
  }
  __syncthreads();

  const int w = tid >> 5;
  const int rt = w >> 2, ct = w & 3;
  const v8f zero = (v8f){0,0,0,0,0,0,0,0};

  v8f r1 = wmma_tile<4>(rbf + rt * 16 * 16, 16, 0, 1, M1, 64, ct * 16, zero);
  store_act_silu(actA + rt * 16 * 64, ct * 16, r1);
  __syncthreads();
  v8f r2 = wmma_tile<16>(actA + rt * 16 * 64, 64, 0, 1, M2, 64, ct * 16, zero);
  store_act_silu(actB + rt * 16 * 64, ct * 16, r2);
  __syncthreads();
  v8f r3 = wmma_tile<16>(actB + rt * 16 * 64, 64, 0, 1, M3, 64, ct * 16, zero);
  store_act_silu(actA + rt * 16 * 64, ct * 16, r3);
  __syncthreads();

  v8f t[4];
#pragma unroll
  for (int j = 0; j < 4; ++j)
    t[j] = wmma_tile<16>(actA + rt * 16 * 64, 64, 0, 1, M4, 256, j * 64 + ct * 16, zero);

  // message math directly on the WMMA D-register layout:
  // lane: channel ch = ct*16 + (lane&15); reg r: edge = e0 + rt*16 + r + 8*(lane>>4)
  const int lane = tid & 31;
  const int ch = ct * 16 + (lane & 15);
  const int half = lane >> 4;
  const float inv3 = 0.57735026918962576f;  // 1/sqrt(3)
#pragma unroll
  for (int r = 0; r < 8; ++r) {
    long e = e0 + rt * 16 + r + half * 8;
    bool valid = e < E;
    long ec = valid ? e : (long)(E - 1);
    int snd = eidx[ec];
    int rcv = eidx[(long)E + ec];
    float sh0 = esh[ec * 4 + 0];
    float shx = esh[ec * 4 + 1];
    float shy = esh[ec * 4 + 2];
    float shz = esh[ec * 4 + 3];
    const float* u = up + (long)snd * 256;
    float x0  = u[ch];
    float x1x = u[64 + ch * 3 + 0];
    float x1y = u[64 + ch * 3 + 1];
    float x1z = u[64 + ch * 3 + 2];
    float dot = x1x * shx + x1y * shy + x1z * shz;
    float m0  = t[0][r] * x0 * sh0 + t[3][r] * dot * inv3;
    float m1x = t[1][r] * x0 * shx + t[2][r] * x1x * sh0;
    float m1y = t[1][r] * x0 * shy + t[2][r] * x1y * sh0;
    float m1z = t[1][r] * x0 * shz + t[2][r] * x1z * sh0;
    if (valid) {
      float* mrow = msg + (long)rcv * 256;
      atomicAdd(mrow + ch, m0);
      atomicAdd(mrow + 64 + ch * 3 + 0, m1x);
      atomicAdd(mrow + 64 + ch * 3 + 1, m1y);
      atomicAdd(mrow + 64 + ch * 3 + 2, m1z);
    }
  }
}

__global__ __launch_bounds__(256) void
poly_kernel(const float* __restrict__ A, const float* __restrict__ Wc0,
            const float* __restrict__ Wc1, float* __restrict__ B, int N) {
  long idx = (long)blockIdx.x * 256 + threadIdx.x;
  if (idx >= (long)N * 64) return;
  long nrow = idx >> 6;
  int c = (int)(idx & 63);
  const float* a = A + nrow * 256;
  float A0 = a[c];
  float ax = a[64 + c * 3 + 0], ay = a[64 + c * 3 + 1], az = a[64 + c * 3 + 2];
  float vv = ax * ax + ay * ay + az * az;
  float A0sq = A0 * A0;
  const float* w0 = Wc0 + c * 5;
  float B0 = w0[0] * A0 + w0[1] * A0sq + w0[2] * vv + w0[3] * A0sq * A0 + w0[4] * A0 * vv;
  const float* w1 = Wc1 + c * 4;
  float s = w1[0] + w1[1] * A0 + w1[2] * A0sq + w1[3] * vv;
  float* b = B + nrow * 256;
  b[c] = B0;
  b[64 + c * 3 + 0] = s * ax;
  b[64 + c * 3 + 1] = s * ay;
  b[64 + c * 3 + 2] = s * az;
}

__global__ __launch_bounds__(256) void
stats_kernel(const float* __restrict__ o, float* __restrict__ stats, int N) {
  __shared__ float red[3 * 256];
  const int tid = threadIdx.x;
  const int c = tid & 63;
  const int sub = tid >> 6;
  long base = (long)blockIdx.x * 1024;
  long end = base + 1024;
  if (end > N) end = N;
  float s0 = 0.f, s1 = 0.f, s2 = 0.f;
  for (long n = base + sub; n < end; n += 4) {
    const float* row = o + n * 256;
    float a = row[c];
    s0 += a;
    s1 += a * a;
    float x = row[64 + c * 3 + 0], y = row[64 + c * 3 + 1], z = row[64 + c * 3 + 2];
    s2 += x * x + y * y + z * z;
  }
  red[tid] = s0; red[256 + tid] = s1; red[512 + tid] = s2;
  __syncthreads();
  if (sub == 0) {
    for (int j = 1; j < 4; ++j) {
      s0 += red[c + 64 * j];
      s1 += red[256 + c + 64 * j];
      s2 += red[512 + c + 64 * j];
    }
    atomicAdd(&stats[c], s0);
    atomicAdd(&stats[64 + c], s1);
    atomicAdd(&stats[128 + c], s2);
  }
}

__global__ __launch_bounds__(256) void
norm_kernel(float* __restrict__ o, const float* __restrict__ stats,
            const float* __restrict__ g0, const float* __restrict__ b0,
            const float* __restrict__ g1, int N) {
  long idx = (long)blockIdx.x * 256 + threadIdx.x;
  if (idx >= (long)N * 64) return;
  long n = idx >> 6;
  int c = (int)(idx & 63);
  float invN = 1.0f / (float)N;
  float mu = stats[c] * invN;
  float var = stats[64 + c] * invN - mu * mu;
  float nrm = stats[128 + c] * invN * (1.0f / 3.0f);
  float* row = o + n * 256;
  row[c] = (row[c] - mu) * rsqrtf(var + 1e-5f) * g0[c] + b0[c];
  float s = rsqrtf(nrm + 1e-5f) * g1[c];
  row[64 + c * 3 + 0] *= s;
  row[64 + c * 3 + 1] *= s;
  row[64 + c * 3 + 2] *= s;
}

extern "C" void kernel_launch(void* const* d_in, const int* in_sizes, int n_in,
                              void* d_out, int out_size, void* d_ws, size_t ws_size,
                              hipStream_t stream) {
  const float* h      = (const float*)d_in[0];
  const int*   eidx   = (const int*)d_in[1];
  const float* esh    = (const float*)d_in[2];
  const float* erbf   = (const float*)d_in[3];
  const float* W_up0  = (const float*)d_in[4];
  const float* W_up1  = (const float*)d_in[5];
  const float* M1     = (const float*)d_in[6];
  const float* M2     = (const float*)d_in[7];
  const float* M3     = (const float*)d_in[8];
  const float* M4     = (const float*)d_in[9];
  const float* W_lin0 = (const float*)d_in[10];
  const float* W_lin1 = (const float*)d_in[11];
  const float* Wsc0   = (const float*)d_in[12];
  const float* Wsc1   = (const float*)d_in[13];
  const float* Wc0    = (const float*)d_in[14];
  const float* Wc1    = (const float*)d_in[15];
  const float* WB0    = (const float*)d_in[16];
  const float* WB1    = (const float*)d_in[17];
  const float* gamma0 = (const float*)d_in[18];
  const float* beta0  = (const float*)d_in[19];
  const float* gamma1 = (const float*)d_in[20];

  const int N = in_sizes[0] / 256;
  const int E = in_sizes[1] / 2;

  float* buf0  = (float*)d_ws;                  // up, later A
  float* buf1  = buf0 + (size_t)N * 256;        // msg, later Bpoly
  float* stats = buf1 + (size_t)N * 256;        // 192 floats
  float* out   = (float*)d_out;

  hipMemsetAsync(buf1, 0, (size_t)N * 256 * sizeof(float), stream);
  hipMemsetAsync(stats, 0, 192 * sizeof(float), stream);

  const int nb = (N + 31) / 32;
  const int eb = (E + 31) / 32;
  const long nc = (long)N * 64;
  const int pb = (int)((nc + 255) / 256);
  const int sb = (N + 1023) / 1024;

  // up = [h0@W_up0 | h1x@W_up1]
  dual_linear_kernel<<<nb, 256, 0, stream>>>(h, W_up0, W_up1, nullptr, nullptr, nullptr, buf0, N);
  // fused edge MLP + gather + message + scatter-add
  edge_kernel<<<eb, 256, 0, stream>>>(eidx, esh, erbf, M1, M2, M3, M4, buf0, buf1, E);
  // A = [msg0@W_lin0 | msg1x@W_lin1]  (overwrites up)
  dual_linear_kernel<<<nb, 256, 0, stream>>>(buf1, W_lin0, W_lin1, nullptr, nullptr, nullptr, buf0, N);
  // B = channelwise polynomial gates (overwrites msg)
  poly_kernel<<<pb, 256, 0, stream>>>(buf0, Wc0, Wc1, buf1, N);
  // o = [B0@WB0 + h0@Wsc0 | B1x@WB1 + h1x@Wsc1] -> d_out
  dual_linear_kernel<<<nb, 256, 0, stream>>>(buf1, WB0, WB1, h, Wsc0, Wsc1, out, N);
  // global channel statistics, then normalize in place
  stats_kernel<<<sb, 256, 0, stream>>>(out, stats, N);
  norm_kernel<<<pb, 256, 0, stream>>>(out, stats, gamma0, beta0, gamma1, N);
}